// MRHG2D_83897891160331
// MI455X (gfx1250) — compile-verified
//
#include <hip/hip_runtime.h>
#include <stdint.h>

typedef float v2f __attribute__((ext_vector_type(2)));
typedef float v4f __attribute__((ext_vector_type(4)));

#define LVLS   16
#define TSIZE  (1u << 19)
#define TMASK  (TSIZE - 1u)
#define BLOCK  256
#define ROWPAD 36            // 32 feature floats + 4 pad -> 16B-aligned rows, tame LDS bank conflicts
#define PRIME_Y 2654435761u

// ---------------------------------------------------------------------------
// Tiny pre-kernel: softmax over the 16 level weights -> d_ws (16 floats).
// ---------------------------------------------------------------------------
__global__ void MRHG2D_softmax16(const float* __restrict__ lw, float* __restrict__ w_out) {
    if (threadIdx.x == 0 && blockIdx.x == 0) {
        float m = lw[0];
        #pragma unroll
        for (int i = 1; i < LVLS; ++i) m = fmaxf(m, lw[i]);
        float e[LVLS];
        float s = 0.f;
        #pragma unroll
        for (int i = 0; i < LVLS; ++i) { e[i] = __expf(lw[i] - m); s += e[i]; }
        const float inv = 1.f / s;
        #pragma unroll
        for (int i = 0; i < LVLS; ++i) w_out[i] = e[i] * inv;
    }
}

// ---------------------------------------------------------------------------
// Main kernel: one thread per point, 16 levels, 4 gathers/level.
//  - positions: NT load (streamed once)
//  - tables:    RT gathers (64 MB total -> resident in 192 MB L2)
//  - output:    staged in LDS, then coalesced NT b128 stores (don't pollute L2)
// ---------------------------------------------------------------------------
__global__ __launch_bounds__(BLOCK) void MRHG2D_hashgrid(
    const float* __restrict__ pos,      // N x 2
    const float* __restrict__ tables,   // L x T x 2
    const float* __restrict__ w,        // 16 softmaxed weights (uniform -> s_load)
    const int*   __restrict__ res,      // 16 resolutions       (uniform -> s_load)
    float* __restrict__ out,            // N x 32
    int N)
{
    __shared__ float lds[BLOCK * ROWPAD];

    const int       tid = threadIdx.x;
    const long long p   = (long long)blockIdx.x * BLOCK + tid;

    if (p < N) {
        // Streamed, read-once position -> non-temporal load (gfx1250 TH_LOAD_NT).
        const v2f xy = __builtin_nontemporal_load((const v2f*)pos + p);

        float* __restrict__ row = &lds[tid * ROWPAD];

        #pragma unroll 4
        for (int l = 0; l < LVLS; ++l) {
            const float rf = (float)res[l];
            const float x  = xy.x * rf;
            const float y  = xy.y * rf;
            const float x0 = floorf(x);
            const float y0 = floorf(y);
            const float fx = x - x0;
            const float fy = y - y0;

            // hash: h = ix ^ (iy * PRIME_Y); T = 2^19 -> mod == AND
            const uint32_t hx0 = (uint32_t)(int)x0;
            const uint32_t hx1 = hx0 + 1u;
            const uint32_t hy0 = (uint32_t)(int)y0 * PRIME_Y;
            const uint32_t hy1 = ((uint32_t)(int)y0 + 1u) * PRIME_Y;

            const uint32_t i00 = (hx0 ^ hy0) & TMASK;
            const uint32_t i10 = (hx1 ^ hy0) & TMASK;
            const uint32_t i01 = (hx0 ^ hy1) & TMASK;
            const uint32_t i11 = (hx1 ^ hy1) & TMASK;

            // Table gathers: default (RT) temporal hint so the 64 MB of tables
            // stay resident in L2 across the whole dispatch.
            const v2f* __restrict__ tab = (const v2f*)tables + (size_t)l * TSIZE;
            const v2f c00 = tab[i00];
            const v2f c10 = tab[i10];
            const v2f c01 = tab[i01];
            const v2f c11 = tab[i11];

            const float gx = 1.f - fx;
            const float gy = 1.f - fy;
            const float w00 = gx * gy;
            const float w10 = fx * gy;
            const float w01 = gx * fy;
            const float w11 = fx * fy;
            const float wl  = w[l];

            v2f f = (c00 * w00 + c10 * w10) + (c01 * w01 + c11 * w11);
            f *= wl;

            // ds_store_b64 into this thread's padded row
            *(v2f*)&row[2 * l] = f;
        }
    }

    __syncthreads();

    // Coalesced write-out: block covers flat region [blockIdx*8192, +8192) elements.
    // 8 rounds of per-lane float4 -> consecutive lanes hit consecutive 16B chunks.
    const long long baseE = (long long)blockIdx.x * (BLOCK * 2 * LVLS);  // elements
    const long long limE  = (long long)N * (2 * LVLS);

    #pragma unroll
    for (int k = 0; k < 8; ++k) {
        const int       idx = k * BLOCK + tid;            // 0..2047 v4-chunks
        const long long e   = baseE + (long long)idx * 4; // element index (16B aligned)
        if (e < limE) {
            const int pl = idx >> 3;          // local point 0..255
            const int ch = (idx & 7) * 4;     // channel 0,4,..,28
            const v4f v  = *(const v4f*)&lds[pl * ROWPAD + ch];   // ds_load_b128
            // Streamed 256 MB output -> non-temporal store (TH_STORE_NT),
            // keeps the hash tables resident in L2.
            __builtin_nontemporal_store(v, (v4f*)out + (e >> 2));
        }
    }
}

extern "C" void kernel_launch(void* const* d_in, const int* in_sizes, int n_in,
                              void* d_out, int out_size, void* d_ws, size_t ws_size,
                              hipStream_t stream) {
    const float* positions   = (const float*)d_in[0];   // N x 2 f32
    const float* tables      = (const float*)d_in[1];   // 16 x 2^19 x 2 f32
    const float* level_wts   = (const float*)d_in[2];   // 16 f32
    const int*   resolutions = (const int*)  d_in[3];   // 16 i32
    float*       out         = (float*)d_out;           // N x 32 f32
    float*       w_soft      = (float*)d_ws;            // 16 f32 scratch

    const int N = in_sizes[0] / 2;

    MRHG2D_softmax16<<<1, 32, 0, stream>>>(level_wts, w_soft);

    const int nblocks = (N + BLOCK - 1) / BLOCK;
    MRHG2D_hashgrid<<<nblocks, BLOCK, 0, stream>>>(positions, tables, w_soft,
                                                   resolutions, out, N);
}